// CaptioningRNN_80891414053413
// MI455X (gfx1250) — compile-verified
//
#include <hip/hip_runtime.h>
#include <math.h>

// ---------------------------------------------------------------------------
// CaptioningRNN (attention LSTM + vocab softmax NLL) for gfx1250 / MI455X.
// - All GEMMs on V_WMMA_F32_16X16X4_F32 (fp32-faithful tensor path).
// - Every wave carries 4 accumulator tiles (M=64 strip) so each B fragment
//   feeds 4 WMMAs (B traffic /4; weights + W_out stay L2-resident: 171MB<192MB).
// - Score GEMM fused with online log-softmax: the 262MB score tensor never
//   touches memory; per-row (max,sum) merged with wave32 shfl_xor.
// - h-tile staged into LDS by the Tensor Data Mover (tensor_load_to_lds +
//   s_wait_tensorcnt), with TDM padding (1 DWORD per 256) -> stride 1028
//   DWORDs -> conflict-free LDS banks for A-fragment reads.
// ---------------------------------------------------------------------------

typedef float v2f __attribute__((ext_vector_type(2)));
typedef float v8f __attribute__((ext_vector_type(8)));
typedef unsigned int v4u __attribute__((ext_vector_type(4)));
typedef int v8i __attribute__((ext_vector_type(8)));
typedef int v4i __attribute__((ext_vector_type(4)));

#define NB   64
#define TT   32
#define DIN  1024
#define WVD  512
#define HID  1024
#define H4   4096
#define VOC  32000
#define NP   16      // spatial positions (4x4)
#define RB   4       // row tiles per wave (M = 64 strip)
#define HS_STRIDE 1028   // 1024 + 4 pad DWORDs (TDM pad_interval=256dw, 1dw)

#if defined(__has_builtin)
#if __has_builtin(__builtin_amdgcn_tensor_load_to_lds)
#define HAVE_TDM 1
#endif
#endif

static __device__ __forceinline__ v8f wmma4(v2f a, v2f b, v8f c) {
  // D = A(16x4,f32) * B(4x16,f32) + C(16x16,f32)
  return __builtin_amdgcn_wmma_f32_16x16x4_f32(
      /*neg_a=*/false, a, /*neg_b=*/false, b,
      /*c_mod=*/(short)0, c, /*reuse_a=*/false, /*reuse_b=*/false);
}
static __device__ __forceinline__ float sigm(float x) {
  return 1.0f / (1.0f + __expf(-x));
}

// ---------------------------------------------------------------------------
// Kernel 1: A[n,p,k] = sum_d feat[n,d,p] * W_proj[d,k] + b_proj[k]
// GEMM M=1024 rows (n,p), N=1024, K=1024.  Wave owns a 64-row strip (4 tiles)
// of one 16-col tile: one B fragment -> 4 WMMAs.
// ---------------------------------------------------------------------------
__global__ void k_proj(const float* __restrict__ feat,
                       const float* __restrict__ Wp,
                       const float* __restrict__ bp,
                       float* __restrict__ As) {
  int tid = threadIdx.x, wv = tid >> 5, lane = tid & 31;
  int job = blockIdx.x * 8 + wv;           // 0..1023 (16 rowgroups x 64 cols)
  int rg = job >> 6, tn = job & 63;
  int r = lane & 15, half = lane >> 4;
  int col = tn * 16 + r;

  v8f acc[RB];
#pragma unroll
  for (int j = 0; j < RB; ++j) acc[j] = (v8f){0.f,0.f,0.f,0.f,0.f,0.f,0.f,0.f};

  for (int kk = 0; kk < DIN; kk += 4) {
    int k0 = kk + half * 2;
    v2f b;
    b.x = Wp[k0 * HID + col];
    b.y = Wp[(k0 + 1) * HID + col];
#pragma unroll
    for (int j = 0; j < RB; ++j) {
      int row = rg * 64 + j * 16 + r;
      int n = row >> 4, p = row & 15;
      v2f a;
      a.x = feat[n * (DIN * NP) + k0 * NP + p];
      a.y = feat[n * (DIN * NP) + (k0 + 1) * NP + p];
      acc[j] = wmma4(a, b, acc[j]);
    }
  }
  float bias = bp[col];
#pragma unroll
  for (int j = 0; j < RB; ++j)
#pragma unroll
    for (int v = 0; v < 8; ++v) {
      int orow = rg * 64 + j * 16 + half * 8 + v;   // C layout: VGPR v -> v / v+8
      As[orow * HID + col] = acc[j][v] + bias;
    }
}

// ---------------------------------------------------------------------------
// Kernel 2: h0 = c0 = mean over p of A[n,p,k]
// ---------------------------------------------------------------------------
__global__ void k_h0(const float* __restrict__ As,
                     float* __restrict__ h_cur, float* __restrict__ c_cur) {
  int idx = blockIdx.x * 256 + threadIdx.x;   // 65536
  int n = idx >> 10, k = idx & 1023;
  float s = 0.f;
  for (int p = 0; p < NP; ++p) s += As[(n * NP + p) * HID + k];
  float h0 = s * (1.0f / 16.0f);
  h_cur[idx] = h0;
  c_cur[idx] = h0;
}

// ---------------------------------------------------------------------------
// Kernel 3a: attention.  s[p] = h . A[n,p,:]/32 ; softmax(16); attn = A^T w
// ---------------------------------------------------------------------------
__global__ void k_attn(const float* __restrict__ As,
                       const float* __restrict__ h_cur,
                       float* __restrict__ attn) {
  __shared__ float sw[NP];
  int n = blockIdx.x;
  int tid = threadIdx.x, wv = tid >> 5, lane = tid & 31;

  for (int p = wv; p < NP; p += 8) {
    float dot = 0.f;
    for (int k = lane; k < HID; k += 32)
      dot += h_cur[n * HID + k] * As[(n * NP + p) * HID + k];
    for (int off = 16; off > 0; off >>= 1) dot += __shfl_xor(dot, off, 32);
    if (lane == 0) sw[p] = dot * (1.0f / 32.0f);   // 1/sqrt(1024)
  }
  __syncthreads();

  float m = -1e30f;
  for (int p = 0; p < NP; ++p) m = fmaxf(m, sw[p]);
  float den = 0.f;
  for (int p = 0; p < NP; ++p) den += __expf(sw[p] - m);
  float inv = 1.0f / den;

  for (int k = tid; k < HID; k += 256) {
    float acc = 0.f;
    for (int p = 0; p < NP; ++p)
      acc += (__expf(sw[p] - m) * inv) * As[(n * NP + p) * HID + k];
    attn[n * HID + k] = acc;
  }
}

// ---------------------------------------------------------------------------
// Kernel 3b: gates a = x_t@Wx + h@Wh + attn@Wattn + b  (64 x 4096)
// Wave owns the full M=64 (4 tiles) of one 16-col tile: B fragment -> 4 WMMAs.
// Embedding rows gathered inline via caption ids.
// ---------------------------------------------------------------------------
__global__ void k_gates(int t,
                        const int*   __restrict__ captions,
                        const float* __restrict__ W_embed,
                        const float* __restrict__ Wx,
                        const float* __restrict__ Wh,
                        const float* __restrict__ Wattn,
                        const float* __restrict__ bvec,
                        const float* __restrict__ h_cur,
                        const float* __restrict__ attn,
                        float* __restrict__ gates) {
  int tid = threadIdx.x, wv = tid >> 5, lane = tid & 31;
  int tn = blockIdx.x * 8 + wv;             // 0..255 column tiles
  int r = lane & 15, half = lane >> 4;
  int col = tn * 16 + r;

  int emb[RB];
#pragma unroll
  for (int j = 0; j < RB; ++j)
    emb[j] = captions[(j * 16 + r) * (TT + 1) + t];   // cap_in[j*16+r, t]

  v8f acc[RB];
#pragma unroll
  for (int j = 0; j < RB; ++j) acc[j] = (v8f){0.f,0.f,0.f,0.f,0.f,0.f,0.f,0.f};

  // x_t @ Wx   (K = 512)
  for (int kk = 0; kk < WVD; kk += 4) {
    int k0 = kk + half * 2;
    v2f b;
    b.x = Wx[k0 * H4 + col];
    b.y = Wx[(k0 + 1) * H4 + col];
#pragma unroll
    for (int j = 0; j < RB; ++j) {
      v2f a;
      a.x = W_embed[emb[j] * WVD + k0];
      a.y = W_embed[emb[j] * WVD + k0 + 1];
      acc[j] = wmma4(a, b, acc[j]);
    }
  }
  // h @ Wh     (K = 1024)
  for (int kk = 0; kk < HID; kk += 4) {
    int k0 = kk + half * 2;
    v2f b;
    b.x = Wh[k0 * H4 + col];
    b.y = Wh[(k0 + 1) * H4 + col];
#pragma unroll
    for (int j = 0; j < RB; ++j) {
      v2f a;
      a.x = h_cur[(j * 16 + r) * HID + k0];
      a.y = h_cur[(j * 16 + r) * HID + k0 + 1];
      acc[j] = wmma4(a, b, acc[j]);
    }
  }
  // attn @ Wattn (K = 1024)
  for (int kk = 0; kk < HID; kk += 4) {
    int k0 = kk + half * 2;
    v2f b;
    b.x = Wattn[k0 * H4 + col];
    b.y = Wattn[(k0 + 1) * H4 + col];
#pragma unroll
    for (int j = 0; j < RB; ++j) {
      v2f a;
      a.x = attn[(j * 16 + r) * HID + k0];
      a.y = attn[(j * 16 + r) * HID + k0 + 1];
      acc[j] = wmma4(a, b, acc[j]);
    }
  }

  float bias = bvec[col];
#pragma unroll
  for (int j = 0; j < RB; ++j)
#pragma unroll
    for (int v = 0; v < 8; ++v) {
      int orow = j * 16 + half * 8 + v;
      gates[orow * H4 + col] = acc[j][v] + bias;
    }
}

// ---------------------------------------------------------------------------
// Kernel 3c: LSTM elementwise
// ---------------------------------------------------------------------------
__global__ void k_lstm(int t, const float* __restrict__ gates,
                       float* __restrict__ h_cur, float* __restrict__ c_cur,
                       float* __restrict__ h_all) {
  int idx = blockIdx.x * 256 + threadIdx.x;  // 65536
  int n = idx >> 10, k = idx & 1023;
  const float* g = gates + n * H4;
  float i = sigm(g[k]);
  float f = sigm(g[HID + k]);
  float o = sigm(g[2 * HID + k]);
  float gg = tanhf(g[3 * HID + k]);
  float c = f * c_cur[idx] + i * gg;
  float h = o * tanhf(c);
  c_cur[idx] = c;
  h_cur[idx] = h;
  h_all[(n * TT + t) * HID + k] = h;
}

// ---------------------------------------------------------------------------
// Kernel 4: target logit   tgt[row] = h_all[row,:] . W_out[:,y] + b_out[y]
// ---------------------------------------------------------------------------
__global__ void k_tgt(const int* __restrict__ captions,
                      const float* __restrict__ h_all,
                      const float* __restrict__ W_out,
                      const float* __restrict__ b_out,
                      float* __restrict__ tgt) {
  int gw = (blockIdx.x * 256 + threadIdx.x) >> 5;  // 0..2047
  int lane = threadIdx.x & 31;
  int n = gw >> 5, t = gw & 31;
  int y = captions[n * (TT + 1) + t + 1];          // cap_out
  float dot = 0.f;
  for (int k = lane; k < HID; k += 32)
    dot += h_all[gw * HID + k] * W_out[(size_t)k * VOC + y];
  for (int off = 16; off > 0; off >>= 1) dot += __shfl_xor(dot, off, 32);
  if (lane == 0) tgt[gw] = dot + b_out[y];
}

// ---------------------------------------------------------------------------
// Kernel 5: fused score GEMM + online logsumexp.
// grid = 32 rowgroups (64 rows) x 8 strips.  64x1024 h-tile staged in LDS by
// the Tensor Data Mover with 1-DWORD pad per 256 (stride 1028 -> bank
// conflict-free A reads).  W_out streamed (L2-resident after first pass).
// One B fragment feeds 4 WMMAs (acc[4]).
// ---------------------------------------------------------------------------
__global__ void k_scores(const float* __restrict__ h_all,
                         const float* __restrict__ W_out,
                         const float* __restrict__ b_out,
                         float* __restrict__ pm, float* __restrict__ ps) {
  extern __shared__ float hs[];                     // 64 * HS_STRIDE floats
  int tid = threadIdx.x, wv = tid >> 5, lane = tid & 31;
  int r = lane & 15, half = lane >> 4;
  int rg = blockIdx.x >> 3;                         // rowgroup 0..31 (64 rows)
  int strip = blockIdx.x & 7;                       // column strip 0..7

#if HAVE_TDM
  if (wv == 0) {
    // Tensor DMA descriptor (ISA ch.8): 2-D tile 1024x64 f32, LDS padding
    // 1 DWORD per 256 DWORDs -> LDS row stride = 1028 DWORDs.
    unsigned lds_off = (unsigned)(size_t)(void*)hs;               // LDS byte addr
    unsigned long long ga =
        (unsigned long long)(const void*)(h_all + (size_t)rg * 64 * HID);
    v4u g0;
    g0.x = 1u;                                                    // count=1
    g0.y = lds_off;                                               // lds_addr
    g0.z = (unsigned)(ga & 0xffffffffu);                          // global[31:0]
    g0.w = (unsigned)((ga >> 32) & 0x1ffffffu) | (2u << 30);      // global[56:32], type=2
    v8i g1;
    g1[0] = (int)((2u << 16)      // data_size = 4B
                | (1u << 20)      // pad_enable
                | (7u << 22));    // pad_interval = 256 DWORDs (pad_amount=1dw)
    g1[1] = 0x04000000;           // tensor_dim0 = 1024 (bits 79:48 low half)
    g1[2] = 0x08000000;           // tensor_dim1 = 2048 (bits 111:80 low half)
    g1[3] = 0x04000000;           // tile_dim0  = 1024 (bits 127:112)
    g1[4] = 64;                   // tile_dim1  = 64 rows
    g1[5] = 1024;                 // tensor_dim0_stride = 1024 (bits 207:160)
    g1[6] = 0x04000000;           // tensor_dim1_stride = 1024 (bits 223:208)
    g1[7] = 0;
    v4i g2 = {0, 0, 0, 0}, g3 = {0, 0, 0, 0};
    v8i g4 = {0, 0, 0, 0, 0, 0, 0, 0};   // 6-arg toolchain variant: extra group
    __builtin_amdgcn_tensor_load_to_lds(g0, g1, g2, g3, g4, 0);
    __builtin_amdgcn_s_wait_tensorcnt(0);
  }
  __syncthreads();
#else
  for (int i = tid; i < 64 * HID; i += 256) {       // same padded layout
    int rr = i >> 10, k = i & 1023;
    hs[rr * HS_STRIDE + k + (k >> 8)] = h_all[(rg * 64 + rr) * HID + k];
  }
  __syncthreads();
#endif

  float m[RB][8], s[RB][8];
#pragma unroll
  for (int j = 0; j < RB; ++j)
#pragma unroll
    for (int v = 0; v < 8; ++v) { m[j][v] = -1e30f; s[j][v] = 0.f; }

  int ct_end = (strip + 1) * 250;                   // 2000/8 tiles per strip

  for (int ct = strip * 250 + wv; ct < ct_end; ct += 8) {
    int col = ct * 16 + r;
    if (ct + 8 < ct_end)                            // global_prefetch next tile
      __builtin_prefetch(&W_out[(size_t)(half * 2) * VOC + (ct + 8) * 16 + r], 0, 0);

    v8f acc[RB];
#pragma unroll
    for (int j = 0; j < RB; ++j) acc[j] = (v8f){0.f,0.f,0.f,0.f,0.f,0.f,0.f,0.f};

    for (int kk = 0; kk < HID; kk += 4) {
      int k0 = kk + half * 2;
      v2f b;
      b.x = W_out[(size_t)k0 * VOC + col];
      b.y = W_out[(size_t)(k0 + 1) * VOC + col];
      int ki = k0 + (kk >> 8);                      // skip intra-row pad DWORDs
#pragma unroll
      for (int j = 0; j < RB; ++j) {
        const float* hrow = hs + (j * 16 + r) * HS_STRIDE;
        v2f a;
        a.x = hrow[ki];
        a.y = hrow[ki + 1];
        acc[j] = wmma4(a, b, acc[j]);
      }
    }
    float bo = b_out[col];
#pragma unroll
    for (int j = 0; j < RB; ++j)
#pragma unroll
      for (int v = 0; v < 8; ++v) {                 // online softmax update
        float x = acc[j][v] + bo;
        float nm = fmaxf(m[j][v], x);
        s[j][v] = s[j][v] * __expf(m[j][v] - nm) + __expf(x - nm);
        m[j][v] = nm;
      }
  }

  // reduce (m,s) across the 16 lanes of each half (rows ...+v / ...+8+v)
  for (int off = 1; off < 16; off <<= 1) {
#pragma unroll
    for (int j = 0; j < RB; ++j)
#pragma unroll
      for (int v = 0; v < 8; ++v) {
        float om = __shfl_xor(m[j][v], off, 32);
        float os = __shfl_xor(s[j][v], off, 32);
        float nm = fmaxf(m[j][v], om);
        s[j][v] = s[j][v] * __expf(m[j][v] - nm) + os * __expf(om - nm);
        m[j][v] = nm;
      }
  }
  if (r == 0) {                                     // lanes 0 and 16
#pragma unroll
    for (int j = 0; j < RB; ++j)
#pragma unroll
      for (int v = 0; v < 8; ++v) {
        int row = rg * 64 + j * 16 + half * 8 + v;
        int slot = row * 64 + strip * 8 + wv;
        pm[slot] = m[j][v];
        ps[slot] = s[j][v];
      }
  }
}

// ---------------------------------------------------------------------------
// Kernel 6: merge 64 (m,s) partials per row -> lse; nll = lse - tgt; masked
// sum / N.  Single block, deterministic.
// ---------------------------------------------------------------------------
__global__ void k_final(const int* __restrict__ captions,
                        const float* __restrict__ tgt,
                        const float* __restrict__ pm,
                        const float* __restrict__ ps,
                        float* __restrict__ out) {
  __shared__ float red[256];
  int tid = threadIdx.x;
  float sum = 0.f;
  for (int row = tid; row < NB * TT; row += 256) {
    float M = -1e30f;
    for (int j = 0; j < 64; ++j) M = fmaxf(M, pm[row * 64 + j]);
    float S = 0.f;
    for (int j = 0; j < 64; ++j) S += ps[row * 64 + j] * __expf(pm[row * 64 + j] - M);
    float lse = M + __logf(S);
    int n = row >> 5, t = row & 31;
    int y = captions[n * (TT + 1) + t + 1];
    if (y != 0) sum += lse - tgt[row];               // NULL_IDX mask
  }
  red[tid] = sum;
  __syncthreads();
  for (int st = 128; st > 0; st >>= 1) {
    if (tid < st) red[tid] += red[tid + st];
    __syncthreads();
  }
  if (tid == 0) out[0] = red[0] * (1.0f / (float)NB);
}

// ---------------------------------------------------------------------------
extern "C" void kernel_launch(void* const* d_in, const int* in_sizes, int n_in,
                              void* d_out, int out_size, void* d_ws, size_t ws_size,
                              hipStream_t stream) {
  (void)in_sizes; (void)n_in; (void)out_size; (void)ws_size;
  const float* feat    = (const float*)d_in[0];
  const int*   caps    = (const int*)  d_in[1];
  const float* W_embed = (const float*)d_in[2];
  const float* Wx      = (const float*)d_in[3];
  const float* Wh      = (const float*)d_in[4];
  const float* Wattn   = (const float*)d_in[5];
  const float* bvec    = (const float*)d_in[6];
  const float* W_proj  = (const float*)d_in[7];
  const float* b_proj  = (const float*)d_in[8];
  const float* W_out   = (const float*)d_in[9];
  const float* b_out   = (const float*)d_in[10];
  float* out = (float*)d_out;

  // workspace layout (floats); total ~15.5 MB
  float* ws    = (float*)d_ws;
  float* As    = ws;                       // 64*16*1024      = 1,048,576
  float* h_cur = As    + 1048576;          // 64*1024
  float* c_cur = h_cur + 65536;
  float* attn  = c_cur + 65536;
  float* gates = attn  + 65536;            // 64*4096         = 262,144
  float* h_all = gates + 262144;           // 64*32*1024      = 2,097,152
  float* tgtb  = h_all + 2097152;          // 2048
  float* pm    = tgtb  + 2048;             // 2048*64
  float* ps    = pm    + 131072;           // 2048*64

  k_proj<<<128, 256, 0, stream>>>(feat, W_proj, b_proj, As);
  k_h0  <<<256, 256, 0, stream>>>(As, h_cur, c_cur);

  for (int t = 0; t < TT; ++t) {
    k_attn <<<64, 256, 0, stream>>>(As, h_cur, attn);
    k_gates<<<32, 256, 0, stream>>>(t, caps, W_embed, Wx, Wh, Wattn, bvec,
                                    h_cur, attn, gates);
    k_lstm <<<256, 256, 0, stream>>>(t, gates, h_cur, c_cur, h_all);
  }

  k_tgt   <<<256, 256, 0, stream>>>(caps, h_all, W_out, b_out, tgtb);
  k_scores<<<256, 256, 64 * HS_STRIDE * sizeof(float), stream>>>(h_all, W_out,
                                                                 b_out, pm, ps);
  k_final <<<1,   256, 0, stream>>>(caps, tgtb, pm, ps, out);
}